// GNNModel_5411658793617
// MI455X (gfx1250) — compile-verified
//
#include <hip/hip_runtime.h>
#include <hip/hip_bf16.h>
#include <math.h>

// ---------------------------------------------------------------------------
// CDNA5 (gfx1250) bipartite GATConv pipeline.
//  - Node GEMMs (x @ W, K=64) run on the matrix core: v_wmma_f32_16x16x4_f32,
//    fp32 in/out (no precision loss), fused with the attention projection
//    alpha = sum_c h[:,h,c] * a[h,c] using wave32 shfl_xor reductions.
//  - Edge softmax + scatter phases are memory/atomic bound; node arrays
//    (<100MB total) are L2-resident on MI455X (192MB L2).
// ---------------------------------------------------------------------------

typedef __attribute__((ext_vector_type(2))) float v2f;
typedef __attribute__((ext_vector_type(8))) float v8f;

// ---------------------------------------------------------------------------
// GEMM + alpha:  H = X[K=64] @ W[64 x NT*16],  AL[n, HEADS] = per-head proj.
// One wave handles a 16-row tile. WMMA f32 16x16x4, 16 k-steps.
//   A frag (16x4 f32): lane<16 -> M=lane, K={4k,4k+1}; lane>=16 -> K={4k+2,4k+3}
//   B frag (4x16 f32): lane<16 -> N=lane, K rows mirrored like A
//   C/D (16x16 f32):   vgpr g: lanes 0-15 -> (M=g, N=lane); 16-31 -> (M=g+8)
// HEADS==NT  : C=16, head == column tile (layer 1, HID=16 matches tile width)
// HEADS==1   : C=NT*16, sum over tiles (layer 2)
// ---------------------------------------------------------------------------
template <int NT, int HEADS>
__global__ __launch_bounds__(256) void gemm_alpha_kernel(
    const float* __restrict__ X, const float* __restrict__ W,
    const float* __restrict__ Avec, float* __restrict__ Hout,
    float* __restrict__ AL, int n_nodes) {
  constexpr int NOUT = NT * 16;
  const int lane = threadIdx.x & 31;
  const int warp = threadIdx.x >> 5;
  const int tile = blockIdx.x * (blockDim.x >> 5) + warp;
  const int row0 = tile * 16;
  if (row0 >= n_nodes) return;  // wave-uniform: EXEC stays all-ones below

  const int half = lane >> 4;       // 0 or 1
  const int l16  = lane & 15;
  const int mrow = row0 + l16;
  const float* xrow = X + (size_t)mrow * 64 + (half << 1);

  v8f acc[NT];
#pragma unroll
  for (int t = 0; t < NT; ++t) acc[t] = (v8f){0.f,0.f,0.f,0.f,0.f,0.f,0.f,0.f};

#pragma unroll 4
  for (int kk = 0; kk < 16; ++kk) {
    const float2 av = *reinterpret_cast<const float2*>(xrow + kk * 4);
    const v2f afrag = {av.x, av.y};
    const int kbase = kk * 4 + (half << 1);
#pragma unroll
    for (int t = 0; t < NT; ++t) {
      const int ncol = t * 16 + l16;
      const v2f bfrag = {W[(size_t)kbase * NOUT + ncol],
                         W[(size_t)(kbase + 1) * NOUT + ncol]};
      acc[t] = __builtin_amdgcn_wmma_f32_16x16x4_f32(
          false, afrag, false, bfrag, (short)0, acc[t], false, false);
    }
  }

  // Store H (source-side features needed by the edge scatter phase).
  if (Hout) {
#pragma unroll
    for (int t = 0; t < NT; ++t)
#pragma unroll
      for (int g = 0; g < 8; ++g)
        Hout[(size_t)(row0 + g + half * 8) * NOUT + t * 16 + l16] = acc[t][g];
  }

  // Attention projection: reduce over the 16 lanes of each half-wave.
#pragma unroll
  for (int g = 0; g < 8; ++g) {
    if (HEADS == NT) {
#pragma unroll
      for (int t = 0; t < NT; ++t) {
        float p = acc[t][g] * Avec[t * 16 + l16];
#pragma unroll
        for (int m = 1; m < 16; m <<= 1) p += __shfl_xor(p, m, 32);
        if (l16 == 0)
          AL[(size_t)(row0 + g + half * 8) * HEADS + t] = p;
      }
    } else {
      float p = 0.f;
#pragma unroll
      for (int t = 0; t < NT; ++t) p += acc[t][g] * Avec[t * 16 + l16];
#pragma unroll
      for (int m = 1; m < 16; m <<= 1) p += __shfl_xor(p, m, 32);
      if (l16 == 0) AL[(size_t)(row0 + g + half * 8)] = p;
    }
  }
}

// ---------------------------------------------------------------------------
// Edge phase helpers
// ---------------------------------------------------------------------------
__device__ __forceinline__ float leaky02(float x) {
  return x > 0.f ? x : 0.2f * x;
}

// Float atomic max via monotonic integer mapping (works with -inf init):
// nonneg floats compare as signed ints; negatives via unsigned min.
__device__ __forceinline__ void atomicMaxF(float* addr, float v) {
  if (v >= 0.f)
    atomicMax(reinterpret_cast<int*>(addr), __float_as_int(v));
  else
    atomicMin(reinterpret_cast<unsigned int*>(addr), __float_as_uint(v));
}

__global__ void edge_max_kernel(const int* __restrict__ src,
                                const int* __restrict__ dst,
                                const float* __restrict__ als,
                                const float* __restrict__ ald,
                                float* __restrict__ mbuf, int nE, int heads) {
  int gid = blockIdx.x * blockDim.x + threadIdx.x;
  if (gid >= nE * heads) return;
  int e = gid / heads, h = gid - e * heads;
  int s = src[e], d = dst[e];
  float lr = leaky02(als[(size_t)s * heads + h] + ald[(size_t)d * heads + h]);
  atomicMaxF(&mbuf[(size_t)d * heads + h], lr);
}

__global__ void edge_expsum_kernel(const int* __restrict__ src,
                                   const int* __restrict__ dst,
                                   const float* __restrict__ als,
                                   const float* __restrict__ ald,
                                   const float* __restrict__ mbuf,
                                   float* __restrict__ den, int nE, int heads) {
  int gid = blockIdx.x * blockDim.x + threadIdx.x;
  if (gid >= nE * heads) return;
  int e = gid / heads, h = gid - e * heads;
  int s = src[e], d = dst[e];
  float lr = leaky02(als[(size_t)s * heads + h] + ald[(size_t)d * heads + h]);
  float ex = __expf(lr - mbuf[(size_t)d * heads + h]);
  atomicAdd(&den[(size_t)d * heads + h], ex);
}

// One thread per (edge, feature): 16/32-thread groups issue coalesced reads
// of hs[src] and same-cacheline f32 atomics into acc[dst].
__global__ void edge_message_kernel(const int* __restrict__ src,
                                    const int* __restrict__ dst,
                                    const float* __restrict__ als,
                                    const float* __restrict__ ald,
                                    const float* __restrict__ mbuf,
                                    const float* __restrict__ den,
                                    const float* __restrict__ hs,
                                    float* __restrict__ acc, int nE, int heads,
                                    int C) {
  const int F = heads * C;
  int gid = blockIdx.x * blockDim.x + threadIdx.x;
  if (gid >= nE * F) return;
  int e = gid / F, f = gid - e * F;
  int h = f / C;
  int s = src[e], d = dst[e];
  float lr = leaky02(als[(size_t)s * heads + h] + ald[(size_t)d * heads + h]);
  float attn = __expf(lr - mbuf[(size_t)d * heads + h]) /
               (den[(size_t)d * heads + h] + 1e-16f);
  atomicAdd(&acc[(size_t)d * F + f], attn * hs[(size_t)s * F + f]);
}

__global__ void fill_kernel(float* __restrict__ p, float v, size_t n) {
  size_t i = (size_t)blockIdx.x * blockDim.x + threadIdx.x;
  if (i < n) p[i] = v;
}

__global__ void finalize_kernel(const float* __restrict__ acc,
                                const float* __restrict__ bias,
                                float* __restrict__ out, int F, size_t n) {
  size_t i = (size_t)blockIdx.x * blockDim.x + threadIdx.x;
  if (i >= n) return;
  float v = acc[i] + bias[i % F];
  out[i] = v > 0.f ? v : 0.f;
}

// ---------------------------------------------------------------------------
// Host orchestration
// ---------------------------------------------------------------------------
namespace {

constexpr int kNDev = 50000, kNTask = 100000, kE = 1000000;

static inline int cdiv(long long a, int b) { return (int)((a + b - 1) / b); }

static void run_conv(hipStream_t stream, const float* x_src,
                     const float* x_dst, int Ns, int Nd, const int* src,
                     const int* dst, const float* Wsrc, const float* Wdst,
                     const float* a_s, const float* a_d, const float* bias,
                     int heads, int C, float* hs, float* als, float* ald,
                     float* mbuf, float* den, float* acc, float* outbuf) {
  const int F = heads * C;
  const int waves_per_block = 8;  // 256 threads, wave32

  // Node GEMM + alpha (source side stores H, dest side alpha-only).
  if (F == 64 && heads == 4) {
    gemm_alpha_kernel<4, 4><<<cdiv(Ns / 16, waves_per_block), 256, 0, stream>>>(
        x_src, Wsrc, a_s, hs, als, Ns);
    gemm_alpha_kernel<4, 4><<<cdiv(Nd / 16, waves_per_block), 256, 0, stream>>>(
        x_dst, Wdst, a_d, nullptr, ald, Nd);
  } else {  // F == 32, heads == 1
    gemm_alpha_kernel<2, 1><<<cdiv(Ns / 16, waves_per_block), 256, 0, stream>>>(
        x_src, Wsrc, a_s, hs, als, Ns);
    gemm_alpha_kernel<2, 1><<<cdiv(Nd / 16, waves_per_block), 256, 0, stream>>>(
        x_dst, Wdst, a_d, nullptr, ald, Nd);
  }

  // Init segment buffers (every call: graph replays do not re-poison).
  fill_kernel<<<cdiv((long long)Nd * heads, 256), 256, 0, stream>>>(
      mbuf, -INFINITY, (size_t)Nd * heads);
  fill_kernel<<<cdiv((long long)Nd * heads, 256), 256, 0, stream>>>(
      den, 0.f, (size_t)Nd * heads);
  fill_kernel<<<cdiv((long long)Nd * F, 256), 256, 0, stream>>>(
      acc, 0.f, (size_t)Nd * F);

  // Softmax-over-incoming-edges in 3 passes.
  edge_max_kernel<<<cdiv((long long)kE * heads, 256), 256, 0, stream>>>(
      src, dst, als, ald, mbuf, kE, heads);
  edge_expsum_kernel<<<cdiv((long long)kE * heads, 256), 256, 0, stream>>>(
      src, dst, als, ald, mbuf, den, kE, heads);
  edge_message_kernel<<<cdiv((long long)kE * F, 256), 256, 0, stream>>>(
      src, dst, als, ald, mbuf, den, hs, acc, kE, heads, C);

  // bias + ReLU into destination buffer.
  finalize_kernel<<<cdiv((long long)Nd * F, 256), 256, 0, stream>>>(
      acc, bias, outbuf, F, (size_t)Nd * F);
}

}  // namespace

extern "C" void kernel_launch(void* const* d_in, const int* in_sizes, int n_in,
                              void* d_out, int out_size, void* d_ws,
                              size_t ws_size, hipStream_t stream) {
  const float* x_dev  = (const float*)d_in[0];
  const float* x_task = (const float*)d_in[1];
  const int* w_src  = (const int*)d_in[2];
  const int* w_dst  = (const int*)d_in[3];
  const int* wb_src = (const int*)d_in[4];
  const int* wb_dst = (const int*)d_in[5];
  const float* W1a_src = (const float*)d_in[6];
  const float* W1a_dst = (const float*)d_in[7];
  const float* a1a_src = (const float*)d_in[8];
  const float* a1a_dst = (const float*)d_in[9];
  const float* b1a     = (const float*)d_in[10];
  const float* W1b_src = (const float*)d_in[11];
  const float* W1b_dst = (const float*)d_in[12];
  const float* a1b_src = (const float*)d_in[13];
  const float* a1b_dst = (const float*)d_in[14];
  const float* b1b     = (const float*)d_in[15];
  const float* W2a_src = (const float*)d_in[16];
  const float* W2a_dst = (const float*)d_in[17];
  const float* a2a_src = (const float*)d_in[18];
  const float* a2a_dst = (const float*)d_in[19];
  const float* b2a     = (const float*)d_in[20];
  const float* W2b_src = (const float*)d_in[21];
  const float* W2b_dst = (const float*)d_in[22];
  const float* a2b_src = (const float*)d_in[23];
  const float* a2b_dst = (const float*)d_in[24];
  const float* b2b     = (const float*)d_in[25];

  // Workspace layout (floats): ~24.0M floats = 96 MB total.
  float* ws   = (float*)d_ws;
  float* hs   = ws;                 // max(Ns)*64      = 6.40M
  float* als  = hs + 6400000;       // 100000*4        = 0.40M
  float* ald  = als + 400000;       //                 = 0.40M
  float* mbuf = ald + 400000;       //                 = 0.40M
  float* den  = mbuf + 400000;      //                 = 0.40M
  float* acc  = den + 400000;       // max(Nd)*64      = 6.40M
  float* xt1  = acc + 6400000;      // x_task_1 100000*64 = 6.40M
  float* xd1  = xt1 + 6400000;      // x_dev_1   50000*64 = 3.20M

  float* x_dev_2  = (float*)d_out;               // 50000*32
  float* x_task_2 = (float*)d_out + kNDev * 32;  // 100000*32

  // Layer 1: dev -> task (heads=4, C=16)
  run_conv(stream, x_dev, x_task, kNDev, kNTask, w_src, w_dst, W1a_src,
           W1a_dst, a1a_src, a1a_dst, b1a, 4, 16, hs, als, ald, mbuf, den, acc,
           xt1);
  // Layer 1: task -> dev
  run_conv(stream, x_task, x_dev, kNTask, kNDev, wb_src, wb_dst, W1b_src,
           W1b_dst, a1b_src, a1b_dst, b1b, 4, 16, hs, als, ald, mbuf, den, acc,
           xd1);
  // Layer 2: dev_1 -> task_1 (heads=1, C=32)
  run_conv(stream, xd1, xt1, kNDev, kNTask, w_src, w_dst, W2a_src, W2a_dst,
           a2a_src, a2a_dst, b2a, 1, 32, hs, als, ald, mbuf, den, acc,
           x_task_2);
  // Layer 2: task_1 -> dev_1
  run_conv(stream, xt1, xd1, kNTask, kNDev, wb_src, wb_dst, W2b_src, W2b_dst,
           a2b_src, a2b_dst, b2b, 1, 32, hs, als, ald, mbuf, den, acc,
           x_dev_2);
}